// QuantumInspiredMemory_26749056319689
// MI455X (gfx1250) — compile-verified
//
#include <hip/hip_runtime.h>
#include <hip/hip_bf16.h>

typedef float v2f __attribute__((ext_vector_type(2)));
typedef float v8f __attribute__((ext_vector_type(8)));

#define NS    64      // memory slots
#define BOND  32
#define NQ    8
#define QDIM  256
#define DIM   1024
#define NB    8       // batch
#define NT    4096    // tokens

// ---------------------------------------------------------------------------
// ws layout (floats):  [0, 8192)   xsum[8][1024]
//                      [8192,16384) rp[8][1024]  (read_proj)
// ---------------------------------------------------------------------------

__global__ void zero_ws(float* __restrict__ p) {
    int i = blockIdx.x * blockDim.x + threadIdx.x;
    if (i < NB * DIM) p[i] = 0.0f;
}

// Partial column sums of x over t, atomically accumulated into xsum[8][1024].
// grid = (8 batches, 4 d-chunks of 256, 32 t-chunks of 128), block = 256.
__global__ void colsum(const float* __restrict__ x, float* __restrict__ xsum) {
    const int b  = blockIdx.x;
    const int d  = blockIdx.y * 256 + threadIdx.x;
    const int t0 = blockIdx.z * 128;
    const float* p = x + ((size_t)b * NT + t0) * DIM + d;
    float s = 0.0f;
#pragma unroll 4
    for (int t = 0; t < 128; ++t) s += p[(size_t)t * DIM];
    atomicAdd(xsum + b * DIM + d, s);
}

// ---------------------------------------------------------------------------
// One block, 1024 threads (32 waves): everything between the two big passes.
// ---------------------------------------------------------------------------
__global__ void __launch_bounds__(1024)
brain(const float* __restrict__ xsum, const float* __restrict__ Wq,
      const float* __restrict__ bq,   const float* __restrict__ Wd,
      const float* __restrict__ bd,   const float* __restrict__ cf,   // [64,2,32]
      const float* __restrict__ cm,   const float* __restrict__ cl,   // [64,6,32,2,32], [64,32,2]
      float* __restrict__ rp) {
    __shared__ float s_xs[NB][DIM];      // x_summary = xsum / T   (32 KB)
    __shared__ float s_qa[NB][QDIM];     // logits -> qa
    __shared__ float s_m0[NB][NQ], s_m1[NB][NQ];
    __shared__ float s_ov[NB][NS];       // overlaps -> attention
    __shared__ float s_feats[NS][NQ];
    __shared__ float s_ro8[NB][NQ];

    const int tid  = threadIdx.x;
    const int wave = tid >> 5;
    const int lane = tid & 31;
    const int M    = lane & 15;          // WMMA row (batch), rows 8..15 are zero pad
    const int Kb   = (lane >> 4) * 2;    // WMMA K sub-offset {0,2}
    const float aMask = (M < NB) ? 1.0f : 0.0f;   // branch-free zero padding
    const int   Mc    = (M < NB) ? M : 0;         // clamped row for padded lanes

    // stage x_summary into LDS + init logits with bias
    {
        const float inv = 1.0f / (float)NT;
        for (int i = tid; i < NB * DIM; i += 1024)
            s_xs[i >> 10][i & (DIM - 1)] = xsum[i] * inv;
        for (int i = tid; i < NB * QDIM; i += 1024)
            s_qa[i >> 8][i & 255] = bq[i & 255];
    }
    __syncthreads();

    // ---- phase 1: logits += x_summary @ Wq via V_WMMA_F32_16X16X4_F32 -----
    {
        const int ntile = wave & 15;         // 16 tiles of 16 columns = 256
        const int khalf = wave >> 4;         // K split across 2 wave groups
        const int n0 = ntile * 16;
        const float* __restrict__ arow = &s_xs[Mc][0];
        v8f acc = {};
#pragma unroll 4
        for (int kk = khalf * 128; kk < khalf * 128 + 128; ++kk) {
            const int K = kk * 4 + Kb;
            v2f a, w;
            a.x = arow[K]     * aMask;
            a.y = arow[K + 1] * aMask;
            const int n = n0 + M;
            w.x = Wq[(size_t)K * QDIM + n];
            w.y = Wq[(size_t)(K + 1) * QDIM + n];
            acc = __builtin_amdgcn_wmma_f32_16x16x4_f32(false, a, false, w,
                                                        (short)0, acc, false, false);
        }
        if (lane < 16) {
            const int n = n0 + lane;
#pragma unroll
            for (int r = 0; r < NB; ++r) atomicAdd(&s_qa[r][n], acc[r]);
        }
    }
    __syncthreads();

    // ---- softmax over QDIM per batch --------------------------------------
    if (tid < NB) {
        float mx = -1e30f;
        for (int n = 0; n < QDIM; ++n) mx = fmaxf(mx, s_qa[tid][n]);
        float sum = 0.0f;
        for (int n = 0; n < QDIM; ++n) { float e = expf(s_qa[tid][n] - mx); s_qa[tid][n] = e; sum += e; }
        const float r = 1.0f / sum;
        for (int n = 0; n < QDIM; ++n) s_qa[tid][n] *= r;
    }
    __syncthreads();

    // ---- amplitude means, with JAX out-of-bounds *clamp* semantics --------
    if (tid < NB * NQ) {
        const int b = tid >> 3, q = tid & 7;
        const int stride = 1 << (NQ - 1 - q);
        float a0 = 0.0f, a1 = 0.0f;
        for (int i = 0; i < stride; ++i) {
            int i0 = i * 2 * stride;          if (i0 > QDIM - 1) i0 = QDIM - 1;
            int i1 = i * 2 * stride + stride; if (i1 > QDIM - 1) i1 = QDIM - 1;
            a0 += s_qa[b][i0];
            a1 += s_qa[b][i1];
        }
        s_m0[b][q] = a0 / (float)stride;
        s_m1[b][q] = a1 / (float)stride;
    }
    __syncthreads();

    // ---- MPS contraction: one (b,s) pair per wave, lane = bond index ------
    for (int pair = wave; pair < NB * NS; pair += 32) {
        const int b = pair >> 6, s = pair & 63;
        float res = s_m0[b][0] * cf[s * 64 + lane] + s_m1[b][0] * cf[s * 64 + 32 + lane];
#pragma unroll
        for (int q = 1; q <= NQ - 2; ++q) {
            const float m0q = s_m0[b][q], m1q = s_m1[b][q];
            const float* bp = cm + (size_t)(s * 6 + (q - 1)) * 2048; // [32][2][32]
            float acc = 0.0f;
#pragma unroll 8
            for (int l = 0; l < 32; ++l) {
                const float rl = __shfl(res, l, 32);
                acc += rl * (m0q * bp[l * 64 + lane] + m1q * bp[l * 64 + 32 + lane]);
            }
            res = acc;
        }
        float v = res * (s_m0[b][7] * cl[s * 64 + 2 * lane]
                       + s_m1[b][7] * cl[s * 64 + 2 * lane + 1]);
#pragma unroll
        for (int off = 16; off; off >>= 1) v += __shfl_down(v, off, 32);
        if (lane == 0) s_ov[b][s] = v;
    }
    __syncthreads();

    // ---- attention softmax (threads 0..7) + slot feats (threads 512..1023)
    if (tid < NB) {
        float mx = -1e30f;
        for (int s = 0; s < NS; ++s) mx = fmaxf(mx, s_ov[tid][s]);
        float sum = 0.0f;
        for (int s = 0; s < NS; ++s) { float e = expf(s_ov[tid][s] - mx); s_ov[tid][s] = e; sum += e; }
        const float r = 1.0f / sum;
        for (int s = 0; s < NS; ++s) s_ov[tid][s] *= r;
    }
    if (tid >= 512) {
        const int t2 = tid - 512;
        const int s = t2 >> 3, j = t2 & 7;
        float f;
        if (j == 0)      { float a = 0; for (int k = 0; k < 64;   ++k) a += cf[s * 64 + k];   f = a / 64.0f; }
        else if (j == 7) { float a = 0; for (int k = 0; k < 64;   ++k) a += cl[s * 64 + k];   f = a / 64.0f; }
        else             { float a = 0; const float* p = cm + (size_t)(s * 6 + j - 1) * 2048;
                           for (int k = 0; k < 2048; ++k) a += p[k];                          f = a / 2048.0f; }
        s_feats[s][j] = f;
    }
    __syncthreads();

    // ---- read_out collapses to 8x8 since slot_values[s][c] = feats[s][c%8]
    if (tid < NB * NQ) {
        const int b = tid >> 3, j = tid & 7;
        float a = 0.0f;
        for (int s = 0; s < NS; ++s) a += s_ov[b][s] * s_feats[s][j];
        s_ro8[b][j] = a;
    }
    __syncthreads();

    // ---- read_proj = read_out @ Wd + bd via WMMA (A[b][c] = ro8[b][c%8]) --
    {
        const float* __restrict__ arow2 = &s_ro8[Mc][0];
        for (int nt = wave; nt < 64; nt += 32) {
            const int n0 = nt * 16;
            v8f acc = {};
#pragma unroll 4
            for (int kk = 0; kk < 64; ++kk) {
                const int K = kk * 4 + Kb;      // K even, K&7 in {0,2,4,6}
                v2f a, w;
                a.x = arow2[K & 7]       * aMask;
                a.y = arow2[(K & 7) + 1] * aMask;
                const int n = n0 + M;
                w.x = Wd[(size_t)K * DIM + n];
                w.y = Wd[(size_t)(K + 1) * DIM + n];
                acc = __builtin_amdgcn_wmma_f32_16x16x4_f32(false, a, false, w,
                                                            (short)0, acc, false, false);
            }
            if (lane < 16) {
                const int n = n0 + lane;
#pragma unroll
                for (int r = 0; r < NB; ++r) rp[r * DIM + n] = acc[r] + bd[n];
            }
        }
    }
}

// ---------------------------------------------------------------------------
// out[b][t][d] = x[b][t][d] + rp[b][d], float4 vectorized.
// ---------------------------------------------------------------------------
__global__ void addbcast(const float4* __restrict__ x, const float4* __restrict__ rp,
                         float4* __restrict__ out) {
    const size_t i = (size_t)blockIdx.x * blockDim.x + threadIdx.x; // over B*T*256 float4s
    const int    d4 = (int)(i & 255);        // DIM/4 = 256
    const int    b  = (int)(i >> 20);        // T*256 = 2^20
    const float4 xv = x[i];
    const float4 rv = rp[(b << 8) + d4];
    float4 o;
    o.x = xv.x + rv.x; o.y = xv.y + rv.y; o.z = xv.z + rv.z; o.w = xv.w + rv.w;
    out[i] = o;
}

extern "C" void kernel_launch(void* const* d_in, const int* in_sizes, int n_in,
                              void* d_out, int out_size, void* d_ws, size_t ws_size,
                              hipStream_t stream) {
    const float* x  = (const float*)d_in[0];
    const float* Wq = (const float*)d_in[1];
    const float* bq = (const float*)d_in[2];
    const float* Wd = (const float*)d_in[3];
    const float* bd = (const float*)d_in[4];
    const float* cf = (const float*)d_in[5];
    const float* cm = (const float*)d_in[6];
    const float* cl = (const float*)d_in[7];
    float* out  = (float*)d_out;
    float* ws   = (float*)d_ws;
    float* xsum = ws;            // 8192 floats
    float* rp   = ws + 8192;     // 8192 floats

    zero_ws<<<32, 256, 0, stream>>>(xsum);
    colsum<<<dim3(8, 4, 32), 256, 0, stream>>>(x, xsum);
    brain<<<1, 1024, 0, stream>>>(xsum, Wq, bq, Wd, bd, cf, cm, cl, rp);

    const size_t total4 = (size_t)NB * NT * (DIM / 4);   // 8,388,608
    addbcast<<<(unsigned)(total4 / 256), 256, 0, stream>>>(
        (const float4*)x, (const float4*)rp, (float4*)out);
    (void)in_sizes; (void)n_in; (void)out_size; (void)ws_size;
}